// CategoricalRegressionLoss_60945585930733
// MI455X (gfx1250) — compile-verified
//
#include <hip/hip_runtime.h>
#include <stdint.h>

typedef float v2f __attribute__((ext_vector_type(2)));
typedef float v8f __attribute__((ext_vector_type(8)));

#define NUM_ATOMS 51
#define V_MIN_F  (-10.0f)
#define V_MAX_F  (10.0f)
#define INV_DZ_F (2.5f)                     /* (51-1)/(10-(-10)) */
#define BS_TOTAL 65536
#define ROWS_TILE 16                        /* rows per wave = one WMMA CE tile */
#define WAVES_PER_BLOCK 8
#define THREADS_PER_BLOCK 256
#define NBLOCKS (BS_TOTAL / (ROWS_TILE * WAVES_PER_BLOCK))   /* 512 */
#define STAGE_F (ROWS_TILE * NUM_ATOMS)     /* 816 contiguous floats per tile */
#define WAVE_LDS_F (2 * STAGE_F + 64)       /* 2 staged arrays + 64 scatter bins */

/* gfx12 DPP16 controls (VALU lane exchange -- no LDS traffic) */
#define DPP_ROW_XMASK1 0x161
#define DPP_ROW_XMASK2 0x162
#define DPP_ROW_XMASK4 0x164
#define DPP_ROW_XMASK8 0x168
#define DPP_ROW_SHARE0 0x150
#define DPP_ROW_SHARE8 0x158

#if __has_builtin(__builtin_amdgcn_wmma_f32_16x16x4_f32)
#define HAVE_WMMA4 1
#else
#define HAVE_WMMA4 0
#warning "missing __builtin_amdgcn_wmma_f32_16x16x4_f32 -- VALU reduce fallback"
#endif

template <int CTRL>
__device__ __forceinline__ float dpp_mov_f32(float v) {
  int r = __builtin_amdgcn_update_dpp(0, __float_as_int(v), CTRL, 0xf, 0xf, true);
  return __int_as_float(r);
}

/* Full-wave (32 lane) reductions: 4 DPP ROW_XMASK butterflies on the VALU,
   one ds_bpermute (__shfl_xor 16) for the cross-half step. */
__device__ __forceinline__ float wave_sum(float v) {
  v += dpp_mov_f32<DPP_ROW_XMASK1>(v);
  v += dpp_mov_f32<DPP_ROW_XMASK2>(v);
  v += dpp_mov_f32<DPP_ROW_XMASK4>(v);
  v += dpp_mov_f32<DPP_ROW_XMASK8>(v);
  v += __shfl_xor(v, 16, 32);
  return v;
}
__device__ __forceinline__ float wave_max(float v) {
  v = fmaxf(v, dpp_mov_f32<DPP_ROW_XMASK1>(v));
  v = fmaxf(v, dpp_mov_f32<DPP_ROW_XMASK2>(v));
  v = fmaxf(v, dpp_mov_f32<DPP_ROW_XMASK4>(v));
  v = fmaxf(v, dpp_mov_f32<DPP_ROW_XMASK8>(v));
  v = fmaxf(v, __shfl_xor(v, 16, 32));
  return v;
}

// CDNA5 async global->LDS copy of STAGE_F consecutive dwords (ASYNCcnt path).
// Low 32 bits of a flat pointer to __shared__ are the LDS offset (ISA 10.2),
// which is what VDST expects.
__device__ __forceinline__ void async_stage(const float* __restrict__ g,
                                            float* l, int lane) {
  #pragma unroll 1
  for (int i = lane; i < STAGE_F; i += 32) {
    uint32_t lds_addr = (uint32_t)(uintptr_t)(l + i);
    uint64_t gaddr    = (uint64_t)(uintptr_t)(g + i);
    asm volatile("global_load_async_to_lds_b32 %0, %1, off"
                 :: "v"(lds_addr), "v"(gaddr) : "memory");
  }
}

__global__ __launch_bounds__(THREADS_PER_BLOCK)
void c51_loss_main(const float* __restrict__ logits_t,
                   const float* __restrict__ logits_tp1,
                   const float* __restrict__ atoms_target,
                   float* __restrict__ block_partials) {
  __shared__ float smem[WAVES_PER_BLOCK * WAVE_LDS_F + WAVES_PER_BLOCK];
  const int tid  = threadIdx.x;
  const int lane = tid & 31;
  const int wid  = tid >> 5;
  float* stage_tp1 = &smem[wid * WAVE_LDS_F];
  float* stage_at  = stage_tp1 + STAGE_F;
  float* bins      = stage_at + STAGE_F;          /* 64 floats, 51 used */
  float* wave_part = &smem[WAVES_PER_BLOCK * WAVE_LDS_F];

  const bool has2 = (lane + 32) < NUM_ATOMS;      /* lanes 0..18 own a 2nd atom */

  const int    gwave = blockIdx.x * WAVES_PER_BLOCK + wid;   /* 0..4095 */
  const size_t base  = (size_t)gwave * STAGE_F;   /* contiguous 816-float block */

  /* Stage softmax source + target atoms via the async engine; prefetch CE logits. */
  async_stage(logits_tp1 + base, stage_tp1, lane);
  async_stage(atoms_target + base, stage_at, lane);
  __builtin_prefetch(logits_t + base + (size_t)lane * 26, 0, 0);
  asm volatile("s_wait_asynccnt 0x0" ::: "memory");

  v2f A; A.x = 0.f; A.y = 0.f;   /* 16x4 f32 WMMA A-tile of CE partials */

  #pragma unroll 1
  for (int rr = 0; rr < ROWS_TILE; ++rr) {
    const int rb = rr * NUM_ATOMS;

    /* ---- softmax(logits_tp1) for this row (2 atoms per lane) ---- */
    float x0 = stage_tp1[rb + lane];
    float x1 = has2 ? stage_tp1[rb + lane + 32] : -3.0e38f;
    float mx = wave_max(fmaxf(x0, x1));
    float e0 = __expf(x0 - mx);
    float e1 = has2 ? __expf(x1 - mx) : 0.f;
    float inv = 1.0f / wave_sum(e0 + e1);
    float p0 = e0 * inv, p1 = e1 * inv;

    /* ---- exact triangular projection: each sample hits exactly 2 bins ---- */
    bins[lane] = 0.f; bins[lane + 32] = 0.f;
    asm volatile("" ::: "memory");
    {
      float tz = fminf(fmaxf(stage_at[rb + lane], V_MIN_F), V_MAX_F);
      float b  = (tz - V_MIN_F) * INV_DZ_F;       /* [0, 50] */
      float fi = floorf(b);
      int   i0 = (int)fi;
      float fr = b - fi;
      atomicAdd(&bins[i0],     p0 * (1.0f - fr));
      atomicAdd(&bins[i0 + 1], p0 * fr);          /* bin 51 only ever gets +0 */
    }
    if (has2) {
      float tz = fminf(fmaxf(stage_at[rb + lane + 32], V_MIN_F), V_MAX_F);
      float b  = (tz - V_MIN_F) * INV_DZ_F;
      float fi = floorf(b);
      int   i0 = (int)fi;
      float fr = b - fi;
      atomicAdd(&bins[i0],     p1 * (1.0f - fr));
      atomicAdd(&bins[i0 + 1], p1 * fr);
    }
    asm volatile("" ::: "memory");

    /* ---- CE vs log_softmax(logits_t) ---- */
    const float* gt = logits_t + base + rb;
    float y0 = gt[lane];
    float y1 = has2 ? gt[lane + 32] : -3.0e38f;
    float my = wave_max(fmaxf(y0, y1));
    float f0 = __expf(y0 - my);
    float f1 = has2 ? __expf(y1 - my) : 0.f;
    float lse = __logf(wave_sum(f0 + f1));
    float pt0 = bins[lane];
    float pt1 = has2 ? bins[lane + 32] : 0.f;
    float part = -(pt0 * ((y0 - my) - lse));
    if (has2) part -= pt1 * ((y1 - my) - lse);

    /* ---- fold 32 lane-partials into A[row=rr, k=0..3] entirely on the VALU:
       3x ROW_XMASK -> 8-lane group sums G0..G3; ROW_SHARE 0/8 broadcasts
       G0/G1 to lanes 0-15 and G2/G3 to lanes 16-31 (WMMA 16x4 A layout:
       lane rr holds K0,K1; lane rr+16 holds K2,K3). ---- */
    float r = part;
    r += dpp_mov_f32<DPP_ROW_XMASK1>(r);
    r += dpp_mov_f32<DPP_ROW_XMASK2>(r);
    r += dpp_mov_f32<DPP_ROW_XMASK4>(r);
    float gA = dpp_mov_f32<DPP_ROW_SHARE0>(r);    /* G0 (rows 0-15) / G2 (16-31) */
    float gB = dpp_mov_f32<DPP_ROW_SHARE8>(r);    /* G1 / G3 */
    if ((lane & 15) == rr) { A.x = gA; A.y = gB; }
  }

  /* ---- CE row sums for all 16 rows in one v_wmma_f32_16x16x4_f32:
     D = A(16x4) * ones(4x16); every column n holds the 16 row sums ---- */
  float wtot;
#if HAVE_WMMA4
  v8f acc = {0.f, 0.f, 0.f, 0.f, 0.f, 0.f, 0.f, 0.f};
  v2f ones; ones.x = 1.0f; ones.y = 1.0f;
  acc = __builtin_amdgcn_wmma_f32_16x16x4_f32(
      /*neg_a=*/false, A, /*neg_b=*/false, ones,
      /*c_mod=*/(short)0, acc, /*reuse_a=*/false, /*reuse_b=*/false);
  float s8 = acc[0] + acc[1] + acc[2] + acc[3] + acc[4] + acc[5] + acc[6] + acc[7];
  wtot = s8 + __shfl_xor(s8, 16, 32);             /* rows 0-7 + rows 8-15 */
#else
  wtot = wave_sum(A.x + A.y);
#endif
  if (lane == 0) wave_part[wid] = wtot;
  __syncthreads();
  if (tid == 0) {
    float bsum = 0.f;
    #pragma unroll
    for (int i = 0; i < WAVES_PER_BLOCK; ++i) bsum += wave_part[i];
    block_partials[blockIdx.x] = bsum;
  }
}

__global__ __launch_bounds__(NBLOCKS)
void c51_loss_reduce(const float* __restrict__ partials, float* __restrict__ out) {
  __shared__ float sm[NBLOCKS];
  const int t = threadIdx.x;
  sm[t] = partials[t];
  __syncthreads();
  for (int s = NBLOCKS / 2; s > 0; s >>= 1) {
    if (t < s) sm[t] += sm[t + s];
    __syncthreads();
  }
  if (t == 0) out[0] = sm[0] * (1.0f / (float)BS_TOTAL);
}

extern "C" void kernel_launch(void* const* d_in, const int* in_sizes, int n_in,
                              void* d_out, int out_size, void* d_ws, size_t ws_size,
                              hipStream_t stream) {
  (void)in_sizes; (void)n_in; (void)out_size; (void)ws_size;
  const float* logits_t   = (const float*)d_in[0];
  const float* logits_tp1 = (const float*)d_in[1];
  const float* atoms_t    = (const float*)d_in[2];
  float* out      = (float*)d_out;
  float* partials = (float*)d_ws;                 /* NBLOCKS floats of scratch */

  c51_loss_main<<<NBLOCKS, THREADS_PER_BLOCK, 0, stream>>>(
      logits_t, logits_tp1, atoms_t, partials);
  c51_loss_reduce<<<1, NBLOCKS, 0, stream>>>(partials, out);
}